// Despawn2D_59983513256498
// MI455X (gfx1250) — compile-verified
//
#include <hip/hip_runtime.h>

typedef __attribute__((ext_vector_type(2))) float v2f;
typedef __attribute__((ext_vector_type(8))) float v8f;

static __device__ __forceinline__ v8f wmma4(v2f a, v2f b, v8f c) {
  // 8 args: (neg_a, A, neg_b, B, c_mod, C, reuse_a, reuse_b)
  return __builtin_amdgcn_wmma_f32_16x16x4_f32(false, a, false, b, (short)0, c,
                                               false, false);
}

// ---------------------------------------------------------------------------
// Prep: materialize the banded-Toeplitz WMMA B operands once per level.
// Table layout per level: [6 chunks][4 filters][32 lanes] of v2f (K-slot pair).
// Filters: 0 = we[k]=w[2k], 1 = wo[k]=w[2k+1], 2 = ge[k]=w[15-2k],
//          3 = go[k]=-w[14-2k].
// Backward synthesis uses the same table: even cols = d*ge + a*we,
// odd cols = -(d*go + a*wo).
// ---------------------------------------------------------------------------
__global__ __launch_bounds__(32)
void despawn_prep(const float* __restrict__ wav, v2f* __restrict__ tab)
{
  const int lane = threadIdx.x;
  const int j  = lane & 15;          // B column (N)
  const int hi = lane >> 4;          // lane-half: K offset 0 or 2
  const int l  = blockIdx.x;         // level
  const float* w = wav + l * 16;
  v2f* t = tab + l * (6 * 4 * 32);

#pragma unroll
  for (int c = 0; c < 6; ++c) {
    const int m0 = c * 4 + hi * 2;   // K slots m0, m0+1
    const int t0 = j + 8 - m0;       // tap index for slot 0
    const int t1 = t0 - 1;           // tap index for slot 1
    const int c0 = min(max(t0, 0), 7);
    const int c1 = min(max(t1, 0), 7);
    const float f0 = ((unsigned)t0 < 8u) ? 1.f : 0.f;  // band mask, branch-free
    const float f1 = ((unsigned)t1 < 8u) ? 1.f : 0.f;
    v2f we = {  w[2 * c0]      * f0,  w[2 * c1]      * f1 };
    v2f wo = {  w[2 * c0 + 1]  * f0,  w[2 * c1 + 1]  * f1 };
    v2f ge = {  w[15 - 2 * c0] * f0,  w[15 - 2 * c1] * f1 };
    v2f go = { -w[14 - 2 * c0] * f0, -w[14 - 2 * c1] * f1 };
    t[(c * 4 + 0) * 32 + lane] = we;
    t[(c * 4 + 1) * 32 + lane] = wo;
    t[(c * 4 + 2) * 32 + lane] = ge;
    t[(c * 4 + 3) * 32 + lane] = go;
  }
}

// ---------------------------------------------------------------------------
// Forward analysis level. One wave per 16x16 output tile, up to `ntiles`
// column tiles per wave with B operands register-resident across tiles.
// ---------------------------------------------------------------------------
__global__ __launch_bounds__(128)
void despawn_fwd_wmma(const float* __restrict__ x, int xstride,
                      float* __restrict__ dout, int dstride,
                      float* __restrict__ aout, int astride,
                      const v2f* __restrict__ bt, int part, int ntiles)
{
  const int lane = threadIdx.x;
  const int j  = lane & 15;
  const int hi = lane >> 4;
  const int rb = (blockIdx.y * 4 + threadIdx.y) * 16;
  const float* xrow = x + (size_t)(rb + j) * (size_t)xstride;

  v2f bwe[6], bwo[6], bge[6], bgo[6];
#pragma unroll
  for (int c = 0; c < 6; ++c) {
    bwe[c] = bt[(c * 4 + 0) * 32 + lane];
    bwo[c] = bt[(c * 4 + 1) * 32 + lane];
    bge[c] = bt[(c * 4 + 2) * 32 + lane];
    bgo[c] = bt[(c * 4 + 3) * 32 + lane];
  }

  for (int tt = 0; tt < ntiles; ++tt) {            // wave-uniform loop
    const int jb = (blockIdx.x * ntiles + tt) * 16;
    const int wbase = jb - 8;
    v8f dacc = {};
    v8f aacc = {};
#pragma unroll
    for (int c = 0; c < 6; ++c) {
      const int m0 = c * 4 + hi * 2;
      int i0 = wbase + m0;     if (i0 < 0) i0 += part;
      int i1 = wbase + m0 + 1; if (i1 < 0) i1 += part;
      const float2 p0 = *(const float2*)(xrow + 2 * i0);  // (even, odd)
      const float2 p1 = *(const float2*)(xrow + 2 * i1);
      v2f ae = {p0.x, p1.x};
      v2f ao = {p0.y, p1.y};
      dacc = wmma4(ae, bwe[c], dacc);
      dacc = wmma4(ao, bwo[c], dacc);
      aacc = wmma4(ae, bge[c], aacc);
      aacc = wmma4(ao, bgo[c], aacc);
    }
    const int orow = rb + 8 * hi;
#pragma unroll
    for (int v = 0; v < 8; ++v) {
      dout[(size_t)(orow + v) * (size_t)dstride + jb + j] = dacc[v];
      aout[(size_t)(orow + v) * (size_t)astride + jb + j] = aacc[v];
    }
  }
}

// ---------------------------------------------------------------------------
// Backward synthesis level:
//   rec[r,2p]   = sum_k d[(p-k)%md]*w[15-2k] + a[(p-k)%md]*w[2k]   (= d*ge + a*we)
//   rec[r,2p+1] = sum_k d[(p-k)%md]*w[14-2k] - a[(p-k)%md]*w[2k+1] (= -(d*go + a*wo))
// Chunk base m0 is even, so the two K-slot gathers collapse to one float2 load
// per array per chunk even after the circular wrap.
// ---------------------------------------------------------------------------
__global__ __launch_bounds__(128)
void despawn_bwd_wmma(const float* __restrict__ d, int dstride,
                      const float* __restrict__ a, int astride,
                      float* __restrict__ rec, int rstride,
                      const v2f* __restrict__ bt, int md, int ntiles)
{
  const int lane = threadIdx.x;
  const int j  = lane & 15;
  const int hi = lane >> 4;
  const int rb = (blockIdx.y * 4 + threadIdx.y) * 16;
  const float* drow = d + (size_t)(rb + j) * (size_t)dstride;
  const float* arow = a + (size_t)(rb + j) * (size_t)astride;

  v2f bwe[6], bwo[6], bge[6], bgo[6];
#pragma unroll
  for (int c = 0; c < 6; ++c) {
    bwe[c] = bt[(c * 4 + 0) * 32 + lane];
    bwo[c] = bt[(c * 4 + 1) * 32 + lane];
    bge[c] = bt[(c * 4 + 2) * 32 + lane];
    bgo[c] = bt[(c * 4 + 3) * 32 + lane];
  }

  for (int tt = 0; tt < ntiles; ++tt) {            // wave-uniform loop
    const int pb = (blockIdx.x * ntiles + tt) * 16;
    const int wbase = pb - 8;                      // even
    v8f eacc = {};
    v8f oacc = {};
#pragma unroll
    for (int c = 0; c < 6; ++c) {
      const int m0 = c * 4 + hi * 2;               // even
      int i0 = wbase + m0; if (i0 < 0) i0 += md;   // pair (i0, i0+1) contiguous
      const float2 pd = *(const float2*)(drow + i0);
      const float2 pa = *(const float2*)(arow + i0);
      v2f ad = {pd.x, pd.y};
      v2f aa = {pa.x, pa.y};
      eacc = wmma4(ad, bge[c], eacc);
      eacc = wmma4(aa, bwe[c], eacc);
      oacc = wmma4(ad, bgo[c], oacc);
      oacc = wmma4(aa, bwo[c], oacc);
    }
    const int orow = rb + 8 * hi;
#pragma unroll
    for (int v = 0; v < 8; ++v) {
      float2 pr;
      pr.x =  eacc[v];                             // rec[2p]
      pr.y = -oacc[v];                             // rec[2p+1]
      *(float2*)(rec + (size_t)(orow + v) * (size_t)rstride + 2 * (pb + j)) = pr;
    }
  }
}

// ---------------------------------------------------------------------------
// Host side: 1 prep + 8 forward + 8 backward launches on `stream`.
// d_out = rec(4096*4096) ++ coeffs(4096*4096); coeff row layout
//   [d0(2048)|d1(1024)|...|d7(16)|approx(16)], off[l] = N - (N>>l).
// d_ws: [0, 32MB) approx ping buffer; [32MB, +48KB) filter operand table.
// The rec region of d_out doubles as the pong buffer (dead until the final
// backward level overwrites it).
// ---------------------------------------------------------------------------
extern "C" void kernel_launch(void* const* d_in, const int* in_sizes, int n_in,
                              void* d_out, int out_size, void* d_ws, size_t ws_size,
                              hipStream_t stream) {
  (void)in_sizes; (void)n_in; (void)out_size; (void)ws_size;
  const int N = 4096;
  const int LEVELS = 8;

  const float* x   = (const float*)d_in[0];
  const float* wav = (const float*)d_in[1];
  float* out   = (float*)d_out;
  float* rec   = out;                                   // 4096*4096
  float* coeff = out + (size_t)N * N;                   // 4096*4096
  float* bufA  = (float*)d_ws;                          // 4096*2048 floats
  v2f*   tab   = (v2f*)((float*)d_ws + (size_t)N * (N / 2));
  float* bufB  = rec;                                   // scratch until last pass

  const dim3 block(32, 4);

  despawn_prep<<<dim3(LEVELS), dim3(32), 0, stream>>>(wav, tab);

  // ---- forward ----
  const float* cur = x;
  int cstride = N;
  for (int l = 0; l < LEVELS; ++l) {
    const int part = N >> (l + 1);
    const int colTiles = part / 16;
    const int ntiles = (colTiles >= 4) ? 4 : colTiles;
    float* dets = coeff + (N - (N >> l));               // details, row stride N
    float* aout;
    int astride;
    if (l == LEVELS - 1) { aout = coeff + (N - 16); astride = N; }  // final approx
    else { aout = (l & 1) ? bufB : bufA; astride = part; }
    dim3 grid(colTiles / ntiles, N / 64);
    despawn_fwd_wmma<<<grid, block, 0, stream>>>(cur, cstride, dets, N,
                                                 aout, astride,
                                                 tab + l * (6 * 4 * 32), part, ntiles);
    cur = aout;
    cstride = astride;
  }

  // ---- backward ----
  const float* arec = coeff + (N - 16);                 // start from final approx
  int arstride = N;
  for (int l = LEVELS - 1; l >= 0; --l) {
    const int md = N >> (l + 1);
    const int pTiles = md / 16;
    const int ntiles = (pTiles >= 4) ? 4 : pTiles;
    const float* dets = coeff + (N - (N >> l));
    float* rout;
    int rstride;
    if (l == 0) { rout = rec; rstride = N; }            // final reconstruction
    else { rout = (l & 1) ? bufA : bufB; rstride = 2 * md; }
    dim3 grid(pTiles / ntiles, N / 64);
    despawn_bwd_wmma<<<grid, block, 0, stream>>>(dets, N, arec, arstride,
                                                 rout, rstride,
                                                 tab + l * (6 * 4 * 32), md, ntiles);
    arec = rout;
    arstride = rstride;
  }
}